// RegressLoss_14731737825397
// MI455X (gfx1250) — compile-verified
//
#include <hip/hip_runtime.h>
#include <hip/hip_bf16.h>

typedef __attribute__((ext_vector_type(2))) float v2f;
typedef __attribute__((ext_vector_type(8))) float v8f;

#define NCH 23
#define HW  4096
#define NUM_CLASS 20

// focal-style BCE with GAMMA=2 (exact square), ALPHA=0.25, EPS=1e-8
__device__ __forceinline__ float focal_loss(float p, float t) {
    float omp = 1.0f - p;
    return -0.25f * t * omp * omp * logf(p + 1e-8f)
           -0.75f * (1.0f - t) * p * p * logf(omp + 1e-8f);
}

// One wave per (batch, 16-pixel tile). 8 waves / 256-thread block, 256 blocks.
// All 8 waves of a block share the same batch (32 blocks per batch), so the
// block cooperatively stages that batch's 4096 targets into LDS once.
//
// The pairwise squared distance is computed ENTIRELY inside the matrix unit:
//   A row M = (-2*px, -2*py, 1, |p|^2)   (K=0,1 in lanes 0-15; K=2,3 in 16-31)
//   B col N = ( tx,    ty,  |t|^2+pen, 1)
//   A.B     = -2(px*tx+py*ty) + |t|^2 + pen + |p|^2 = d^2 (+ invalid penalty)
__global__ __launch_bounds__(256) void regress_loss_main(
    const float* __restrict__ pred, const float* __restrict__ lab,
    float* __restrict__ blockPart)
{
    __shared__ float sTx [HW];     // target x, indexed by target n
    __shared__ float sTy [HW];     // target y
    __shared__ float sAdd[HW];     // |t|^2 + (invalid ? 1e30 : 0)
    __shared__ float sMin[8 * 16];
    __shared__ float sRed[8][6];

    const int wid   = threadIdx.x >> 5;
    const int lane  = threadIdx.x & 31;
    const int tile  = blockIdx.x * 8 + wid;   // 0..2047
    const int b     = tile >> 8;              // 0..7 (uniform within block)
    const int tileM = tile & 255;             // 0..255 (pixel tile within batch)

    const float* predB = pred + (size_t)b * NCH * HW;
    const float* labB  = lab  + (size_t)b * NCH * HW;
    const float* prdX  = predB + 21 * HW;
    const float* prdY  = predB + 22 * HW;
    const float* labX  = labB  + 21 * HW;
    const float* labY  = labB  + 22 * HW;

    // ---- Stage this batch's targets into LDS (once per block) ----
    for (int k = 0; k < 16; ++k) {
        int n  = k * 256 + threadIdx.x;
        // tpts: transpose(0,3,2,1) -> target n reads plane index (n%64)*64 + n/64
        int tr = ((n & 63) << 6) | (n >> 6);
        float tx = labX[tr], ty = labY[tr];
        // valid: computed in [H,W] layout -> direct flat index n (faithful to source)
        float vx = labX[n], vy = labY[n];
        float va = (fmaxf(vx, vy) > 0.01f) ? 0.0f : 1.0e30f;
        sTx[n]  = tx;
        sTy[n]  = ty;
        sAdd[n] = tx * tx + ty * ty + va;
    }
    __syncthreads();

    // ---------------- Phase A: pairwise min squared-distance via WMMA ----------------
    const bool loHalf = (lane < 16);
    float npv = 0.0f;
    float ax = 0.0f, ay = 0.0f;
    if (loHalf) {
        int pix = tileM * 16 + lane;          // pixel = h*64 + w (row-major H,W)
        ax = prdX[pix]; ay = prdY[pix];
        npv = ax * ax + ay * ay;              // |p|^2 for M=lane
    }
    // lanes 16-31 need |p|^2 of M = lane-16 for their K=3 slot
    float npq = __shfl(npv, lane & 15, 32);
    v2f avec;                                  // A tile (16x4 f32)
    avec.x = loHalf ? (-2.0f * ax) : 1.0f;     // K=0 : K=2
    avec.y = loHalf ? (-2.0f * ay) : npq;      // K=1 : K=3

    float minacc[8];
    #pragma unroll
    for (int j = 0; j < 8; ++j) minacc[j] = 3.0e38f;

    for (int tn = 0; tn < 256; ++tn) {
        // All 32 lanes read column data for N = lane&15 (LDS broadcast across halves,
        // distinct banks within a half): loop stays free of EXEC divergence.
        int nl = tn * 16 + (lane & 15);
        float tx   = sTx [nl];
        float ty   = sTy [nl];
        float addq = sAdd[nl];                 // |t|^2 + invalid penalty for col N
        v2f bvec;                              // B tile (4x16 f32)
        bvec.x = loHalf ? tx : addq;           // K=0 : K=2
        bvec.y = loHalf ? ty : 1.0f;           // K=1 : K=3
        // D[M][N] = d^2(M,N)  (C/D layout: VGPR j -> M=j+8*(lane>=16), N=lane&15)
        v8f d2 = __builtin_amdgcn_wmma_f32_16x16x4_f32(
            false, avec, false, bvec, (short)0, (v8f){}, false, false);
        #pragma unroll
        for (int j = 0; j < 8; ++j)
            minacc[j] = fminf(minacc[j], d2[j]);
    }
    // min across the 16 lanes of each half (xor masks 1..8 stay within a half)
    #pragma unroll
    for (int j = 0; j < 8; ++j) {
        #pragma unroll
        for (int m = 8; m >= 1; m >>= 1)
            minacc[j] = fminf(minacc[j], __shfl_xor(minacc[j], m, 32));
    }
    const int off = (lane >> 4) << 3;          // 0 for lanes 0-15, 8 for lanes 16-31
    if ((lane & 15) == 0) {
        #pragma unroll
        for (int j = 0; j < 8; ++j)
            sMin[wid * 16 + off + j] = minacc[j];   // lane0 -> M=0..7, lane16 -> M=8..15
    }
    __syncthreads();

    // ---------------- Phase B: per-pixel losses + partial reduction ----------------
    float s1 = 0.0f, s2 = 0.0f, s3 = 0.0f, s4 = 0.0f, s5 = 0.0f, s6 = 0.0f;
    if (loHalf) {
        int pix = tileM * 16 + lane;
        float minv = sMin[wid * 16 + lane];
        float tp = labB[pix];                 // t_prob (channel 0)
        float pp = predB[pix];                // p_prob
        float probLoss = focal_loss(pp, tp);
        float dx = prdX[pix] - labX[pix];
        float dy = prdY[pix] - labY[pix];
        float dloss = (dx * dx + dy * dy) * 0.0625f;   // / (RATIO/2)^2 = /16
        float ign = (minv >= 0.25f) ? 1.0f : 0.0f;     // d >= 0.5  <=>  d^2 >= 0.25
        float omt = 1.0f - tp;
        for (int c = 1; c <= NUM_CLASS; ++c) {
            float pc = predB[c * HW + pix];
            float tc = labB [c * HW + pix];
            float cl = focal_loss(pc, tc);
            s3 += cl * tp;
            s4 += cl * omt;
        }
        s1 = probLoss * tp;
        s2 = probLoss * omt * ign;
        s5 = dloss * tp;
        s6 = (tp > 0.5f) ? 1.0f : 0.0f;
    }
    #pragma unroll
    for (int m = 16; m >= 1; m >>= 1) {
        s1 += __shfl_xor(s1, m, 32);
        s2 += __shfl_xor(s2, m, 32);
        s3 += __shfl_xor(s3, m, 32);
        s4 += __shfl_xor(s4, m, 32);
        s5 += __shfl_xor(s5, m, 32);
        s6 += __shfl_xor(s6, m, 32);
    }
    if (lane == 0) {
        sRed[wid][0] = s1; sRed[wid][1] = s2; sRed[wid][2] = s3;
        sRed[wid][3] = s4; sRed[wid][4] = s5; sRed[wid][5] = s6;
    }
    __syncthreads();
    if (threadIdx.x == 0) {
        float acc[6] = {0.0f, 0.0f, 0.0f, 0.0f, 0.0f, 0.0f};
        for (int w = 0; w < 8; ++w)
            #pragma unroll
            for (int k = 0; k < 6; ++k) acc[k] += sRed[w][k];
        #pragma unroll
        for (int k = 0; k < 6; ++k) blockPart[blockIdx.x * 6 + k] = acc[k];
    }
}

// Deterministic final fold of 256 block partials -> 5 output scalars.
__global__ __launch_bounds__(32) void regress_loss_final(
    const float* __restrict__ part, float* __restrict__ out)
{
    int lane = threadIdx.x;
    float s[6] = {0.0f, 0.0f, 0.0f, 0.0f, 0.0f, 0.0f};
    for (int blk = lane; blk < 256; blk += 32) {
        #pragma unroll
        for (int k = 0; k < 6; ++k) s[k] += part[blk * 6 + k];
    }
    #pragma unroll
    for (int k = 0; k < 6; ++k) {
        #pragma unroll
        for (int m = 16; m >= 1; m >>= 1)
            s[k] += __shfl_xor(s[k], m, 32);
    }
    if (lane == 0) {
        float P = fmaxf(s[5], 1.0f);                       // positive_sum
        float loss_coord = s[4] / P * 5.0f;                // W_COORD
        float loss_obj   = s[0] / P;                       // W_OBJ = 1
        float loss_noobj = s[1] * (0.5f / 32768.0f);       // mean over B*H*W, * W_NOOBJ
        float loss_class = s[2] / P + s[3] * (0.5f / 655360.0f); // + mean/2
        out[0] = loss_coord + loss_obj + loss_noobj + loss_class;
        out[1] = loss_coord;
        out[2] = loss_obj;
        out[3] = loss_noobj;
        out[4] = loss_class;
    }
}

extern "C" void kernel_launch(void* const* d_in, const int* in_sizes, int n_in,
                              void* d_out, int out_size, void* d_ws, size_t ws_size,
                              hipStream_t stream) {
    const float* pred = (const float*)d_in[0];
    const float* lab  = (const float*)d_in[1];
    float* out = (float*)d_out;
    float* blockPart = (float*)d_ws;   // 256 blocks * 6 partials = 6 KB

    regress_loss_main<<<256, 256, 0, stream>>>(pred, lab, blockPart);
    regress_loss_final<<<1, 32, 0, stream>>>(blockPart, out);
}